// MaskedDiffusion_36644660969934
// MI455X (gfx1250) — compile-verified
//
#include <hip/hip_runtime.h>

typedef __attribute__((ext_vector_type(2))) float v2f;
typedef __attribute__((ext_vector_type(8))) float v8f;

#define NCAT 1024
#define BB   16
#define LL   4096
#define MASK_IDX 1023
#define EPSF 1e-8f

__device__ __forceinline__ float wave_max32(float m) {
#pragma unroll
    for (int off = 16; off >= 1; off >>= 1)
        m = fmaxf(m, __shfl_xor(m, off, 32));
    return m;
}

__device__ __forceinline__ float wave_sum32(float s) {
#pragma unroll
    for (int off = 16; off >= 1; off >>= 1)
        s += __shfl_xor(s, off, 32);
    return s;
}

// Kernel 1: lse[r] = logsumexp(W[r, :]) for r in [0, 1024).
// One wave owns 16 rows. Pass A: coalesced row max. Pass B: exp + WMMA row-sum
// (B = ones -> every column of D is the row sum), f32 accumulation on the matrix pipe.
__global__ __launch_bounds__(128) void lse_rows_kernel(const float* __restrict__ W,
                                                       float* __restrict__ lse) {
    const int lane    = threadIdx.x & 31;
    const int wib     = threadIdx.x >> 5;        // wave in block: 0..3
    const int wave    = blockIdx.x * 4 + wib;    // 0..63
    const int rowBase = wave * 16;

    __shared__ float smax[64];

    // ---- Pass A: per-row max (coalesced b128 loads, full-wave reduce per row) ----
    float mxr[16];
#pragma unroll
    for (int r = 0; r < 16; ++r) {
        const float4* rowp = (const float4*)(W + (size_t)(rowBase + r) * NCAT);
        float m = -3.402823466e38f;
#pragma unroll
        for (int it = 0; it < 8; ++it) {         // 32 lanes * 4 floats * 8 = 1024 cols
            float4 v = rowp[it * 32 + lane];
            m = fmaxf(m, fmaxf(fmaxf(v.x, v.y), fmaxf(v.z, v.w)));
        }
        mxr[r] = wave_max32(m);                  // wave-uniform row max
    }
    if (lane == 0) {
#pragma unroll
        for (int r = 0; r < 16; ++r) smax[wib * 16 + r] = mxr[r];
    }
    __syncthreads();

    // ---- Pass B: acc = sum_k exp(W[row,k] - rowmax) via V_WMMA_F32_16X16X4_F32 ----
    // 32-bit A 16x4 layout: lanes 0-15 hold {K=0,K=1} of row=lane; lanes 16-31 hold {K=2,K=3}.
    const int    r     = lane & 15;
    const float  mymax = smax[wib * 16 + r];
    const float* rowp  = W + (size_t)(rowBase + r) * NCAT + ((lane >> 4) << 1);

    v8f acc = {};
    v2f bones = {1.0f, 1.0f};                    // B (4x16) = all ones, layout-independent
    for (int k = 0; k < NCAT; k += 4) {
        float2 w2 = *(const float2*)(rowp + k);  // b64 load: 2 consecutive cols
        v2f a;
        a[0] = __expf(w2.x - mymax);
        a[1] = __expf(w2.y - mymax);
        // 8 args: (neg_a, A, neg_b, B, c_mod, C, reuse_a, reuse_b)
        acc = __builtin_amdgcn_wmma_f32_16x16x4_f32(false, a, false, bones,
                                                    (short)0, acc, false, false);
    }

    // D layout: VGPR i -> row rowBase+i on lanes 0-15, row rowBase+8+i on lanes 16-31.
    float hi[8];
#pragma unroll
    for (int i = 0; i < 8; ++i) hi[i] = __shfl(acc[i], 16, 32);
    if (lane == 0) {
#pragma unroll
        for (int i = 0; i < 8; ++i) {
            lse[rowBase + i]     = __logf(acc[i]) + mxr[i];
            lse[rowBase + 8 + i] = __logf(hi[i])  + mxr[8 + i];
        }
    }
}

// Kernel 2: per-position loss term via gather; deterministic block reduction -> partial[256].
__global__ __launch_bounds__(256) void loss_kernel(const int* __restrict__ x,
                                                   const float* __restrict__ u,
                                                   const float* __restrict__ noise,
                                                   const float* __restrict__ W,
                                                   const float* __restrict__ lse,
                                                   float* __restrict__ partial) {
    const int idx = blockIdx.x * 256 + threadIdx.x;  // 0 .. B*L-1 (grid exactly covers)
    const int b   = idx >> 12;                       // L = 4096

    const float uu = u[0];
    float t = uu + (float)b * (1.0f / 16.0f);
    t = t - floorf(t);                               // remainder(u + b/B, 1)
    const float alpha = 1.0f - t;
    const float wgt   = -1.0f / fmaxf(t, EPSF);      // alpha'(t)/clamp(1-alpha) = -1/clamp(t)

    const int   xv = x[idx];
    const float nv = noise[idx];
    const int   z  = (nv < alpha) ? xv : MASK_IDX;   // keep w.p. alpha else mask
    const float term = W[(size_t)z * NCAT + xv] - lse[z];  // log_softmax gathered at x
    float val = wgt * term * (1.0f / (float)BB);     // fold batch-mean in

    val = wave_sum32(val);
    __shared__ float ssum[8];
    const int lane = threadIdx.x & 31;
    const int wib  = threadIdx.x >> 5;
    if (lane == 0) ssum[wib] = val;
    __syncthreads();
    if (threadIdx.x < 32) {
        float v = (lane < 8) ? ssum[lane] : 0.0f;
#pragma unroll
        for (int off = 4; off >= 1; off >>= 1) v += __shfl_xor(v, off, 32);
        if (lane == 0) partial[blockIdx.x] = v;
    }
}

// Kernel 3: reduce 256 partials -> scalar output (deterministic).
__global__ __launch_bounds__(256) void final_reduce_kernel(const float* __restrict__ partial,
                                                           float* __restrict__ out) {
    float v = partial[threadIdx.x];
    v = wave_sum32(v);
    __shared__ float ssum[8];
    const int lane = threadIdx.x & 31;
    const int wib  = threadIdx.x >> 5;
    if (lane == 0) ssum[wib] = v;
    __syncthreads();
    if (threadIdx.x < 32) {
        float s = (lane < 8) ? ssum[lane] : 0.0f;
#pragma unroll
        for (int off = 4; off >= 1; off >>= 1) s += __shfl_xor(s, off, 32);
        if (lane == 0) out[0] = s;
    }
}

extern "C" void kernel_launch(void* const* d_in, const int* in_sizes, int n_in,
                              void* d_out, int out_size, void* d_ws, size_t ws_size,
                              hipStream_t stream) {
    (void)in_sizes; (void)n_in; (void)out_size; (void)ws_size;
    const int*   x     = (const int*)d_in[0];     // [B, L] int32
    const float* u     = (const float*)d_in[1];   // [1]
    const float* noise = (const float*)d_in[2];   // [B, L]
    const float* W     = (const float*)d_in[3];   // [N, N]

    float* lse     = (float*)d_ws;                // 1024 floats
    float* partial = lse + NCAT;                  // 256 floats
    float* out     = (float*)d_out;

    lse_rows_kernel<<<16, 128, 0, stream>>>(W, lse);                         // 64 waves
    loss_kernel<<<(BB * LL) / 256, 256, 0, stream>>>(x, u, noise, W, lse, partial);
    final_reduce_kernel<<<1, 256, 0, stream>>>(partial, out);
}